// Attention_17386027614758
// MI455X (gfx1250) — compile-verified
//
#include <hip/hip_runtime.h>
#include <hip/hip_bf16.h>

typedef __attribute__((ext_vector_type(2))) float v2f;
typedef __attribute__((ext_vector_type(8))) float v8f;
typedef __attribute__((ext_vector_type(4))) unsigned int u32x4;
typedef __attribute__((ext_vector_type(8))) int i32x8;
typedef __attribute__((ext_vector_type(4))) int i32x4;

#define S_TOTAL 8192
#define DDIM    256
#define NSLOT   64
#define STILE   64          // s-rows per block
#define LPAD    260         // 256 + 4 pad -> conflict-free 16-lane row fetch
#define LSTR    65          // 64 + 1 pad for logits rows
#define IDX_SLOT 3
#define NEGV   (-9.0e15f)

#if defined(__has_builtin)
#if __has_builtin(__builtin_amdgcn_tensor_load_to_lds) && \
    __has_builtin(__builtin_amdgcn_s_wait_tensorcnt)
#define USE_TDM 1
#endif
#endif

#ifdef USE_TDM
typedef __attribute__((address_space(3))) void lds_void;
__device__ static inline unsigned lds_byte_off(void* p) {
  // generic -> LDS addrspace cast; AS(3) pointers are raw LDS byte offsets
  return (unsigned)(unsigned long long)(lds_void*)p;
}

// D# group1 for a 64-row x 256-dword tile, row stride 256 dwords in memory,
// data_size=4B, LDS padding: +4 dwords every 256 dwords  => LPAD=260 layout.
__device__ static inline i32x8 tdm_group1_64x256() {
  i32x8 g1;
  g1[0] = (2 << 16)      // data_size = 4 bytes
        | (1 << 20)      // pad_enable
        | (7 << 22)      // pad_interval: 256 dwords
        | (3 << 25);     // pad_amount:   4 dwords
  g1[1] = (256 & 0xFFFF) << 16;  // tensor_dim0 low16 (=256 elements)
  g1[2] = (64 & 0xFFFF) << 16;   // tensor_dim0 hi16=0 | tensor_dim1 low16 (=64)
  g1[3] = (256 & 0xFFFF) << 16;  // tensor_dim1 hi16=0 | tile_dim0 (=256)
  g1[4] = 64;                    // tile_dim1 (=64 rows), tile_dim2=0
  g1[5] = 256;                   // tensor_dim0_stride low32 (=256 elements)
  g1[6] = 0;                     // dim0_stride hi16 | dim1_stride low16
  g1[7] = 0;
  return g1;
}

__device__ static inline u32x4 tdm_group0(unsigned lds_addr, const void* gp) {
  const unsigned long long ga = (unsigned long long)gp;
  u32x4 g0;
  g0[0] = 1u;                                  // count=1 (valid descriptor)
  g0[1] = lds_addr;                            // lds_addr bytes
  g0[2] = (unsigned)(ga & 0xFFFFFFFFu);        // global_addr[31:0]
  g0[3] = ((unsigned)(ga >> 32) & 0x01FFFFFFu) // global_addr[56:32]
        | 0x80000000u;                         // type=2 ("image")
  return g0;
}
#endif

// ---------------- K0: qk[n,d] = (queries @ Wk)[n,d]/16 ; qb[n] = (queries[n]·bk)/16
__global__ void __launch_bounds__(256)
prep_qk_kernel(const float* __restrict__ q, const float* __restrict__ Wk,
               const float* __restrict__ bk, float* __restrict__ qk,
               float* __restrict__ qb) {
  const int n = blockIdx.x;   // 0..63
  const int d = threadIdx.x;  // 0..255
  float acc = 0.f;
  for (int e = 0; e < DDIM; ++e)
    acc = fmaf(q[n * DDIM + e], Wk[e * DDIM + d], acc);
  qk[n * DDIM + d] = acc * 0.0625f;   // fold 1/sqrt(256)
  if (d == 0) {
    float a = 0.f;
    for (int e = 0; e < DDIM; ++e) a = fmaf(q[n * DDIM + e], bk[e], a);
    qb[n] = a * 0.0625f;
  }
}

// ---------------- K1: main pass. One block = 64 s-rows, 4 waves, each wave 16 rows x 64 slots.
__global__ void __launch_bounds__(128)
attn_pass1_kernel(const float* __restrict__ gIn, const int* __restrict__ gMask,
                  const float* __restrict__ qk, const float* __restrict__ qb,
                  float* __restrict__ uOut, float* __restrict__ Pblk,
                  float* __restrict__ Zblk) {
  extern __shared__ float smem[];
  float* inS = smem;                     // 64*260
  float* qkS = inS + STILE * LPAD;       // 64*260
  float* lS  = qkS + NSLOT * LPAD;       // 64*65
  float* uS  = lS + STILE * LSTR;        // 64
  float* qbS = uS + STILE;               // 64

  const int tid   = threadIdx.x;
  const int blk   = blockIdx.x;          // 0..2047
  const int b     = blk >> 7;            // /128 tiles per batch
  const int sBase = (blk & 127) * STILE;

  const float* inTile = gIn + ((size_t)b * S_TOTAL + sBase) * DDIM;
  const int*   mBase  = gMask + (size_t)b * NSLOT * S_TOTAL + sBase;

  // CDNA5 prefetch of the mask tile we will need after the GEMM
  __builtin_prefetch(mBase, 0, 1);

#ifdef USE_TDM
  // ---- Tensor Data Mover: DMA both operand tiles straight into padded LDS.
  if (tid < 32) {  // wave 0 issues; EXEC is ignored by TENSOR_LOAD_TO_LDS
    const i32x8 g1 = tdm_group1_64x256();
    const i32x4 gz4 = {0, 0, 0, 0};
    const i32x8 gz8 = {0, 0, 0, 0, 0, 0, 0, 0};   // unused VADDR4 slot
    __builtin_amdgcn_tensor_load_to_lds(
        tdm_group0(lds_byte_off(inS), inTile), g1, gz4, gz4, gz8, 0);
    __builtin_amdgcn_tensor_load_to_lds(
        tdm_group0(lds_byte_off(qkS), qk), g1, gz4, gz4, gz8, 0);
    __builtin_amdgcn_s_wait_tensorcnt(0);
  }
  if (tid < NSLOT) qbS[tid] = qb[tid];
  __syncthreads();
#else
  // ---- cooperative stage: input tile (64x256) and qk (64x256) into padded LDS
  for (int i = tid; i < (STILE * DDIM) / 4; i += 128) {
    const int row = i >> 6;       // (i*4)/256
    const int c4  = i & 63;
    const float4 v = reinterpret_cast<const float4*>(inTile)[i];
    *reinterpret_cast<float4*>(&inS[row * LPAD + c4 * 4]) = v;
  }
  for (int i = tid; i < (NSLOT * DDIM) / 4; i += 128) {
    const int row = i >> 6;
    const int c4  = i & 63;
    const float4 v = reinterpret_cast<const float4*>(qk)[i];
    *reinterpret_cast<float4*>(&qkS[row * LPAD + c4 * 4]) = v;
  }
  if (tid < NSLOT) qbS[tid] = qb[tid];
  __syncthreads();
#endif

  // ---- WMMA: logits tile [16 s-rows x 64 slots] per wave, K = 256 in steps of 4
  const int wave = tid >> 5;     // wave32
  const int lane = tid & 31;
  const int hh   = lane >> 4;    // 0: K pair 0,1   1: K pair 2,3
  const int l16  = lane & 15;
  const int aRow = wave * 16 + l16;

  v8f acc0 = {}, acc1 = {}, acc2 = {}, acc3 = {};
  for (int kk = 0; kk < DDIM / 4; ++kk) {
    const int kb = kk * 4 + 2 * hh;
    const v2f aV = *reinterpret_cast<const v2f*>(&inS[aRow * LPAD + kb]);
    const v2f b0 = *reinterpret_cast<const v2f*>(&qkS[( 0 + l16) * LPAD + kb]);
    const v2f b1 = *reinterpret_cast<const v2f*>(&qkS[(16 + l16) * LPAD + kb]);
    const v2f b2 = *reinterpret_cast<const v2f*>(&qkS[(32 + l16) * LPAD + kb]);
    const v2f b3 = *reinterpret_cast<const v2f*>(&qkS[(48 + l16) * LPAD + kb]);
    acc0 = __builtin_amdgcn_wmma_f32_16x16x4_f32(false, aV, false, b0, (short)0, acc0, false, false);
    acc1 = __builtin_amdgcn_wmma_f32_16x16x4_f32(false, aV, false, b1, (short)0, acc1, false, false);
    acc2 = __builtin_amdgcn_wmma_f32_16x16x4_f32(false, aV, false, b2, (short)0, acc2, false, false);
    acc3 = __builtin_amdgcn_wmma_f32_16x16x4_f32(false, aV, false, b3, (short)0, acc3, false, false);
  }

  // ---- C/D layout: VGPR r holds M=r (lanes 0-15) / M=r+8 (lanes 16-31), N=lane%16
#pragma unroll
  for (int r = 0; r < 8; ++r) {
    const int row = wave * 16 + r + 8 * hh;
    lS[row * LSTR +  0 + l16] = acc0[r];
    lS[row * LSTR + 16 + l16] = acc1[r];
    lS[row * LSTR + 32 + l16] = acc2[r];
    lS[row * LSTR + 48 + l16] = acc3[r];
  }
  __syncthreads();

  // ---- per-(b,s): mask, softmax over slots, pick slot IDX
  if (tid < STILE) {
    const int* mrow = mBase + tid;  // mask[b][n][sBase+tid], stride S_TOTAL over n
    float m = -3.4e38f;
    for (int n = 0; n < NSLOT; ++n) {
      float x = lS[tid * LSTR + n] + qbS[n];
      if (mrow[(size_t)n * S_TOTAL] == 0) x = NEGV;
      lS[tid * LSTR + n] = x;
      m = fmaxf(m, x);
    }
    float ssum = 0.f, xi = 0.f;
    for (int n = 0; n < NSLOT; ++n) {
      const float e = __expf(lS[tid * LSTR + n] - m);
      ssum += e;
      if (n == IDX_SLOT) xi = e;
    }
    const float u = xi / ssum;
    uS[tid] = u;
    uOut[(size_t)b * S_TOTAL + sBase + tid] = u;
  }
  __syncthreads();

  // ---- per-block partials: P[e] = sum_s u[s]*x[s,e] ; Z = sum_s u[s]
  for (int e = tid; e < DDIM; e += 128) {
    float acc = 0.f;
    for (int s = 0; s < STILE; ++s)
      acc = fmaf(uS[s], inS[s * LPAD + e], acc);
    Pblk[(size_t)blk * DDIM + e] = acc;
  }
  if (tid == 0) {
    float z = 0.f;
    for (int s = 0; s < STILE; ++s) z += uS[s];
    Zblk[blk] = z;
  }
}

// ---------------- K2: reduce partials per batch, apply Wv^T + bv
__global__ void __launch_bounds__(256)
finalize_kernel(const float* __restrict__ Pblk, const float* __restrict__ Zblk,
                const float* __restrict__ Wv, const float* __restrict__ bv,
                float* __restrict__ out, float* __restrict__ Zfin) {
  __shared__ float sP[DDIM];
  __shared__ float sZ[128];
  const int b = blockIdx.x;     // 0..15
  const int t = threadIdx.x;    // 0..255

  float p = 0.f;
  for (int k = 0; k < 128; ++k)
    p += Pblk[((size_t)(b * 128 + k)) * DDIM + t];
  sP[t] = p;
  if (t < 128) sZ[t] = Zblk[b * 128 + t];
  __syncthreads();
  for (int off = 64; off > 0; off >>= 1) {
    if (t < off) sZ[t] += sZ[t + off];
    __syncthreads();
  }
  const float invZ = 1.f / sZ[0];
  float dot = 0.f;
  for (int e = 0; e < DDIM; ++e)
    dot = fmaf(sP[e], Wv[t * DDIM + e], dot);   // values = x @ Wv^T
  out[b * DDIM + t] = bv[t] + invZ * dot;
  if (t == 0) Zfin[b] = sZ[0];
}

// ---------------- K3: attention output = u / Z[b]
__global__ void __launch_bounds__(256)
scale_u_kernel(const float* __restrict__ u, const float* __restrict__ Zfin,
               float* __restrict__ att) {
  const int i = blockIdx.x * 256 + threadIdx.x;   // 0..131071
  const int b = i >> 13;                          // /8192
  att[i] = u[i] / Zfin[b];
}

extern "C" void kernel_launch(void* const* d_in, const int* in_sizes, int n_in,
                              void* d_out, int out_size, void* d_ws, size_t ws_size,
                              hipStream_t stream) {
  const float* input  = (const float*)d_in[0];   // [16,8192,256]
  const int*   mask   = (const int*)  d_in[1];   // [16,64,8192]
  const float* Wv     = (const float*)d_in[2];   // [256,256]
  const float* bv     = (const float*)d_in[3];   // [256]
  const float* Wk     = (const float*)d_in[4];   // [256,256]
  const float* bk     = (const float*)d_in[5];   // [256]
  const float* qry    = (const float*)d_in[6];   // [64,256]

  float* out = (float*)d_out;                    // [16*256] then [16*8192]
  float* att = out + 16 * DDIM;

  // workspace carve-up (floats)
  float* ws   = (float*)d_ws;
  float* qk   = ws;                      // 64*256
  float* qb   = qk + NSLOT * DDIM;       // 64
  float* u    = qb + NSLOT;              // 16*8192
  float* Pblk = u + 16 * S_TOTAL;        // 2048*256
  float* Zblk = Pblk + 2048 * DDIM;      // 2048
  float* Zfin = Zblk + 2048;             // 16

  const size_t smem1 =
      (size_t)(STILE * LPAD + NSLOT * LPAD + STILE * LSTR + STILE + NSLOT) *
      sizeof(float);  // ~147 KB dynamic LDS

  prep_qk_kernel<<<NSLOT, 256, 0, stream>>>(qry, Wk, bk, qk, qb);
  attn_pass1_kernel<<<2048, 128, smem1, stream>>>(input, mask, qk, qb, u, Pblk, Zblk);
  finalize_kernel<<<16, 256, 0, stream>>>(Pblk, Zblk, Wv, bv, out, Zfin);
  scale_u_kernel<<<512, 256, 0, stream>>>(u, Zfin, att);
}